// PlainFCOS_73040213835838
// MI455X (gfx1250) — compile-verified
//
#include <hip/hip_runtime.h>
#include <stdint.h>

#define NBINS     8192      // top-13-bit histogram of orderable float
#define BIN_SHIFT 19
#define CAP       8192      // candidate cap (expected ~2-3K for k=1000)
#define SORT_N    8192      // bitonic sort size (64KB LDS of u64 keys)
#define CHUNK     8192      // floats per TDM tile (32KB)

typedef __attribute__((ext_vector_type(4))) uint32_t u32x4;
typedef __attribute__((ext_vector_type(8))) uint32_t u32x8;
typedef __attribute__((ext_vector_type(4))) int      i32x4;
typedef __attribute__((ext_vector_type(8))) int      i32x8;

// ---- order-preserving float <-> uint transform (larger float => larger uint)
__device__ __forceinline__ uint32_t f2ord(uint32_t b) {
    uint32_t mask = (uint32_t)((int32_t)b >> 31) | 0x80000000u;
    return b ^ mask;
}
__device__ __forceinline__ uint32_t ord2bits(uint32_t u) {
    return (u & 0x80000000u) ? (u ^ 0x80000000u) : ~u;
}

// ---- TDM: issue one tensor_load_to_lds of CHUNK floats starting at `start`
// D# group0: [0]=count=1, [1]=lds byte addr, [2..3]=57-bit global addr + type=2
// D# group1: data_size=4B, tensor = rem x 1, tile = CHUNK x 1
__device__ __forceinline__ void tdm_issue(const float* __restrict__ x,
                                          size_t start, size_t n,
                                          uint32_t ldsoff) {
    uint64_t ga    = (uint64_t)(uintptr_t)x + start * 4ull;
    size_t   remsz = n - start;
    uint32_t rem   = (remsz > 0xFFFFFFFFull) ? 0xFFFFFFFFu : (uint32_t)remsz;

    u32x4 g0;
    g0[0] = 1u;                                                // count=1 (valid user D#)
    g0[1] = ldsoff;                                            // lds_addr (bytes)
    g0[2] = (uint32_t)ga;                                      // global_addr[31:0]
    g0[3] = ((uint32_t)(ga >> 32) & 0x01FFFFFFu) | (2u << 30); // addr[56:32] | type=2

    u32x8 g1;
    g1[0] = 2u << 16;                 // workgroup_mask=0, data_size=2 (4 bytes)
    g1[1] = (rem & 0xFFFFu) << 16;    // tensor_dim0[15:0] @ bits 63:48
    g1[2] = (rem >> 16) | (1u << 16); // tensor_dim0[31:16] | tensor_dim1[15:0]=1
    g1[3] = (uint32_t)CHUNK << 16;    // tensor_dim1[31:16]=0 | tile_dim0=CHUNK
    g1[4] = 1u;                       // tile_dim1 = 1, tile_dim2 = 0
    g1[5] = rem;                      // tensor_dim0_stride[31:0]
    g1[6] = 0u;                       // stride0[47:32], stride1[15:0]
    g1[7] = 0u;                       // stride1[47:16]

#if __has_builtin(__builtin_amdgcn_tensor_load_to_lds_d2)
    __builtin_amdgcn_tensor_load_to_lds_d2((i32x4)g0, (i32x8)g1, 0);
#else
    // amdgpu-toolchain (clang-23) 6-arg form:
    // (uint32x4 g0, int32x8 g1, int32x4 g2, int32x4 g3, int32x8, i32 cpol)
    i32x4 z4 = (i32x4)0;
    i32x8 z8 = (i32x8)0;
    __builtin_amdgcn_tensor_load_to_lds(g0, (i32x8)g1, z4, z4, z8, 0);
#endif
}

// ---- kernel 0: zero histogram + meta (runs every call => deterministic replay)
__global__ void fcos_zero_kernel(uint32_t* __restrict__ hist,
                                 uint32_t* __restrict__ meta) {
    int i = blockIdx.x * blockDim.x + threadIdx.x;
    if (i < NBINS) hist[i] = 0u;
    if (i < 4)     meta[i] = 0u;   // meta[0]=counter, meta[1]=threshold
}

// ---- kernel 1: streaming histogram via TDM double-buffer (320MB, BW-bound)
__global__ __launch_bounds__(256)
void fcos_hist_kernel(const float* __restrict__ x, size_t n,
                      uint32_t* __restrict__ ghist) {
    __shared__ uint32_t lh[NBINS];                 // 32KB bins
    __shared__ float    buf[2][CHUNK];             // 2 x 32KB tile buffers

    for (int i = threadIdx.x; i < NBINS; i += blockDim.x) lh[i] = 0u;
    __syncthreads();

    const size_t nchunks = (n + CHUNK - 1) / CHUNK;
    const size_t cstride = gridDim.x;
    const uint32_t lds0 = (uint32_t)(uintptr_t)&buf[0][0];
    const uint32_t lds1 = (uint32_t)(uintptr_t)&buf[1][0];

    size_t c = blockIdx.x;
    if (threadIdx.x < 32) {                        // wave 0 only (TDM ignores EXEC)
        if (c < nchunks) tdm_issue(x, c * CHUNK, n, lds0);
    }
    int pb = 0;
    for (; c < nchunks; c += cstride) {
        if (threadIdx.x < 32) {
            size_t cn = c + cstride;
            if (cn < nchunks) {                    // prefetch next tile into other buffer
                tdm_issue(x, cn * CHUNK, n, pb ? lds0 : lds1);
                __builtin_amdgcn_s_wait_tensorcnt(1);   // oldest tile (c) complete
            } else {
                __builtin_amdgcn_s_wait_tensorcnt(0);
            }
        }
        __syncthreads();                           // tile visible to all 8 waves

        size_t start = c * CHUNK;
        size_t remsz = n - start;
        int lim  = (remsz < (size_t)CHUNK) ? (int)remsz : CHUNK;
        int lim4 = lim >> 2;
        const float4* b4 = (const float4*)&buf[pb][0];
        for (int i = threadIdx.x; i < lim4; i += 256) {
            float4 v = b4[i];                      // ds_load_b128
            atomicAdd(&lh[f2ord(__float_as_uint(v.x)) >> BIN_SHIFT], 1u);
            atomicAdd(&lh[f2ord(__float_as_uint(v.y)) >> BIN_SHIFT], 1u);
            atomicAdd(&lh[f2ord(__float_as_uint(v.z)) >> BIN_SHIFT], 1u);
            atomicAdd(&lh[f2ord(__float_as_uint(v.w)) >> BIN_SHIFT], 1u);
        }
        for (int i = (lim4 << 2) + threadIdx.x; i < lim; i += 256) {
            atomicAdd(&lh[f2ord(__float_as_uint(buf[pb][i])) >> BIN_SHIFT], 1u);
        }
        __syncthreads();                           // all done reading buf[pb]
        pb ^= 1;
    }

    for (int i = threadIdx.x; i < NBINS; i += blockDim.x) {
        uint32_t h = lh[i];
        if (h) atomicAdd(&ghist[i], h);
    }
}

// ---- kernel 2: find threshold bin (k-th largest lives in bin T)
__global__ void fcos_thresh_kernel(const uint32_t* __restrict__ ghist,
                                   const int* __restrict__ topk_p,
                                   uint32_t* __restrict__ meta) {
    if (threadIdx.x == 0) {
        uint64_t k   = (uint64_t)(*topk_p);
        uint64_t acc = 0;
        int T = 0;
        for (int b = NBINS - 1; b >= 0; --b) {
            acc += ghist[b];
            if (acc >= k) { T = b; break; }
        }
        meta[1] = (uint32_t)T << BIN_SHIFT;   // orderable threshold (bin floor)
    }
}

// ---- kernel 3: compact candidates >= threshold into packed u64 keys
// key = (ord_value << 32) | ~index  => descending sort == jax.lax.top_k order
__global__ void fcos_compact_kernel(const float4* __restrict__ x, size_t n4,
                                    const float* __restrict__ xs, size_t n,
                                    const uint32_t* __restrict__ meta_ro,
                                    uint32_t* __restrict__ meta,
                                    uint64_t* __restrict__ cand) {
    uint32_t thr = meta_ro[1];
    size_t stride = (size_t)gridDim.x * blockDim.x;
    size_t gtid   = (size_t)blockIdx.x * blockDim.x + threadIdx.x;
    for (size_t i = gtid; i < n4; i += stride) {
        __builtin_prefetch(&x[i + stride], 0, 0);   // global_prefetch_b8
        float4 v = x[i];                            // global_load_b128
        uint32_t u0 = f2ord(__float_as_uint(v.x));
        uint32_t u1 = f2ord(__float_as_uint(v.y));
        uint32_t u2 = f2ord(__float_as_uint(v.z));
        uint32_t u3 = f2ord(__float_as_uint(v.w));
        uint32_t bi = (uint32_t)(i * 4);
        if (u0 >= thr) { uint32_t p = atomicAdd(&meta[0], 1u);
            if (p < CAP) cand[p] = ((uint64_t)u0 << 32) | (uint32_t)~(bi + 0u); }
        if (u1 >= thr) { uint32_t p = atomicAdd(&meta[0], 1u);
            if (p < CAP) cand[p] = ((uint64_t)u1 << 32) | (uint32_t)~(bi + 1u); }
        if (u2 >= thr) { uint32_t p = atomicAdd(&meta[0], 1u);
            if (p < CAP) cand[p] = ((uint64_t)u2 << 32) | (uint32_t)~(bi + 2u); }
        if (u3 >= thr) { uint32_t p = atomicAdd(&meta[0], 1u);
            if (p < CAP) cand[p] = ((uint64_t)u3 << 32) | (uint32_t)~(bi + 3u); }
    }
    size_t base = n4 * 4;
    if (gtid < (n - base)) {
        uint32_t u = f2ord(__float_as_uint(xs[base + gtid]));
        if (u >= thr) { uint32_t p = atomicAdd(&meta[0], 1u);
            if (p < CAP) cand[p] = ((uint64_t)u << 32) | (uint32_t)~((uint32_t)(base + gtid)); }
    }
}

// ---- kernel 4: single-WGP bitonic sort of candidates + final outputs
__global__ __launch_bounds__(1024)
void fcos_final_kernel(const float* __restrict__ box,
                       const uint64_t* __restrict__ cand,
                       const uint32_t* __restrict__ meta,
                       const int* __restrict__ topk_p,
                       const int* __restrict__ w_p,
                       const int* __restrict__ h_p,
                       float* __restrict__ out, int nc) {
    __shared__ uint64_t keys[SORT_N];   // 64KB of the 320KB/WGP LDS
    const int tid = threadIdx.x;
    const int nt  = blockDim.x;

    uint32_t M = meta[0];
    if (M > CAP) M = CAP;
    for (int i = tid; i < SORT_N; i += nt)
        keys[i] = (i < (int)M) ? cand[i] : 0ull;   // pad sorts to bottom
    __syncthreads();

    // bitonic sort, descending
    for (int kk = 2; kk <= SORT_N; kk <<= 1) {
        for (int j = kk >> 1; j > 0; j >>= 1) {
            for (int i = tid; i < SORT_N; i += nt) {
                int ixj = i ^ j;
                if (ixj > i) {
                    uint64_t a = keys[i], b = keys[ixj];
                    bool desc = ((i & kk) == 0);
                    if (desc ? (a < b) : (a > b)) { keys[i] = b; keys[ixj] = a; }
                }
            }
            __syncthreads();
        }
    }

    const int   k  = *topk_p;
    const float fw = (float)(*w_p);
    const float fh = (float)(*h_p);
    float* ob = out;                  // [k,4] normalized boxes
    float* os = out + 4 * k;          // [k]   scores
    int*   ol = (int*)(out + 5 * k);  // [k]   int32 labels

    for (int r = tid; r < k; r += nt) {
        uint64_t key = keys[r];
        uint32_t u   = (uint32_t)(key >> 32);
        uint32_t idx = ~(uint32_t)(key & 0xFFFFFFFFull);
        float    val = __uint_as_float(ord2bits(u));
        os[r] = 1.0f / (1.0f + __expf(-val));
        int bi  = (int)(idx / (uint32_t)nc);
        int lab = (int)(idx - (uint32_t)bi * (uint32_t)nc);
        ol[r] = lab;
        const float* bp = box + (size_t)bi * 4;
        float x0 = bp[0] / fw, y0 = bp[1] / fh, x1 = bp[2] / fw, y1 = bp[3] / fh;
        ob[4 * r + 0] = fminf(fmaxf(x0, 0.0f), 1.0f);
        ob[4 * r + 1] = fminf(fmaxf(y0, 0.0f), 1.0f);
        ob[4 * r + 2] = fminf(fmaxf(x1, 0.0f), 1.0f);
        ob[4 * r + 3] = fminf(fmaxf(y1, 0.0f), 1.0f);
    }
}

extern "C" void kernel_launch(void* const* d_in, const int* in_sizes, int n_in,
                              void* d_out, int out_size, void* d_ws, size_t ws_size,
                              hipStream_t stream) {
    (void)n_in; (void)out_size; (void)ws_size;
    const float*  cls    = (const float*)d_in[0];   // [1,Nq,Nc] logits
    const float*  box    = (const float*)d_in[1];   // [1,Nq,4]
    const int*    topk_p = (const int*)d_in[2];
    const int*    w_p    = (const int*)d_in[3];
    const int*    h_p    = (const int*)d_in[4];

    size_t n  = (size_t)in_sizes[0];                // Nq*Nc
    int    nq = in_sizes[1] / 4;
    int    nc = (int)(n / (size_t)nq);
    size_t n4 = n / 4;

    // workspace layout
    uint32_t* hist = (uint32_t*)d_ws;                               // 32KB
    uint32_t* meta = hist + NBINS;                                  // 16B
    uint64_t* cand = (uint64_t*)((char*)d_ws + 40 * 1024);          // 64KB

    const int TPB  = 256;
    const int GRID = 2048;

    fcos_zero_kernel<<<(NBINS + TPB - 1) / TPB, TPB, 0, stream>>>(hist, meta);
    fcos_hist_kernel<<<GRID, TPB, 0, stream>>>(cls, n, hist);
    fcos_thresh_kernel<<<1, 64, 0, stream>>>(hist, topk_p, meta);
    fcos_compact_kernel<<<GRID, TPB, 0, stream>>>((const float4*)cls, n4, cls, n,
                                                  meta, meta, cand);
    fcos_final_kernel<<<1, 1024, 0, stream>>>(box, cand, meta, topk_p, w_p, h_p,
                                              (float*)d_out, nc);
}